// LiftSplatShootNet_44873818308900
// MI455X (gfx1250) — compile-verified
//
#include <hip/hip_runtime.h>

// ---------------- problem constants (from reference) ----------------
#define B_    4
#define N_    6
#define D_    41
#define CAMC_ 64
#define FH_   16
#define FW_   44
#define NX_   200
#define NY_   200
#define NCH   105            // D + CAMC
#define HW    (FH_*FW_)      // 704
#define NPIX  (B_*N_*FH_*FW_)// 16896 = 2112 * 8
#define CELLS (B_*NX_*NY_)   // 160000 (NZ == 1)

typedef float v2f __attribute__((ext_vector_type(2)));
typedef float v8f __attribute__((ext_vector_type(8)));

// ---------------- helpers ----------------
__device__ __forceinline__ void inv3(const float* __restrict__ m, float* __restrict__ o) {
    float a=m[0],b=m[1],c=m[2],d=m[3],e=m[4],f=m[5],g=m[6],h=m[7],i=m[8];
    float A =  e*i - f*h;
    float Bc = -(d*i - f*g);
    float Cc =  d*h - e*g;
    float r = 1.0f / (a*A + b*Bc + c*Cc);
    o[0]=A*r;  o[1]=(c*h-b*i)*r; o[2]=(b*f-c*e)*r;
    o[3]=Bc*r; o[4]=(a*i-c*g)*r; o[5]=(c*d-a*f)*r;
    o[6]=Cc*r; o[7]=(b*g-a*h)*r; o[8]=(a*e-b*d)*r;
}

__device__ __forceinline__ float wave_max32(float v) {
    #pragma unroll
    for (int o = 16; o > 0; o >>= 1) v = fmaxf(v, __shfl_xor(v, o, 32));
    return v;
}
__device__ __forceinline__ float wave_sum32(float v) {
    #pragma unroll
    for (int o = 16; o > 0; o >>= 1) v += __shfl_xor(v, o, 32);
    return v;
}

// ---------------- kernel 0: zero the accumulator ----------------
__global__ void lss_zero(float4* __restrict__ p, long n4) {
    long i = (long)blockIdx.x * blockDim.x + threadIdx.x;
    long stride = (long)gridDim.x * blockDim.x;
    float4 z = {0.f, 0.f, 0.f, 0.f};
    for (; i < n4; i += stride) p[i] = z;
}

// ---------------- kernel 1: softmax + geometry + WMMA lift + scatter ----------------
// One wave32 per camera pixel, 8 waves/block. Entirely LDS-free: all cross-lane
// exchange is shfl/bpermute; control flow is wave-uniform so EXEC is all-ones at
// every WMMA site. WS=true: channel-contiguous accumulator with dump cell
// (unconditional atomics). WS=false: direct output layout, guarded atomics.
template <bool WS>
__global__ __launch_bounds__(256) void lss_main(
    const float* __restrict__ xf,      // (B,N,105,16,44)
    const float* __restrict__ rots,    // (B,N,3,3)
    const float* __restrict__ trans,   // (B,N,3)
    const float* __restrict__ intrins, // (B,N,3,3)
    const float* __restrict__ post_rots,
    const float* __restrict__ post_trans,
    float* __restrict__ acc)
{
    const int lane = threadIdx.x & 31;
    const int pid  = blockIdx.x * 8 + (threadIdx.x >> 5);

    const int w = pid % FW_;
    const int h = (pid / FW_) % FH_;
    const int n = (pid / HW) % N_;
    const int b = pid / (HW * N_);
    const int bn = b * N_ + n;

    const float* pix = xf + ((long)bn * NCH * FH_ + h) * (long)FW_ + w;

    // ---- softmax over 41 depth logits (lane owns bin `lane`, plus bin 32+lane if lane<9)
    float l0 = pix[(long)lane * HW];
    float l1 = (lane < 9) ? pix[(long)(32 + lane) * HW] : -3.4e38f;
    float mx = wave_max32(fmaxf(l0, l1));
    float e0 = __expf(l0 - mx);
    float e1 = (lane < 9) ? __expf(l1 - mx) : 0.0f;
    float inv_s = 1.0f / wave_sum32(e0 + e1);
    const float depth0 = e0 * inv_s;   // bins  0..31 (bin = lane)
    const float depth1 = e1 * inv_s;   // bins 32..40 (bin = 32+lane, lane<9), else exact 0

    // ---- feature channels, loaded directly in WMMA-B lane order:
    // lane l (<16) holds channels l, 16+l, 32+l, 48+l; hi lanes contribute zeros.
    const bool hi   = lane >= 16;
    const int  cOff = lane & 15;
    float fb[4];
    #pragma unroll
    for (int u = 0; u < 4; ++u)
        fb[u] = hi ? 0.0f : pix[(long)(D_ + u * 16 + cOff) * HW];

    // ---- per-(b,n) matrices (wave-uniform, ~100 redundant flops)
    float iPR[9], iK[9], C[9];
    inv3(post_rots + bn * 9, iPR);
    inv3(intrins  + bn * 9, iK);
    const float* R = rots + bn * 9;
    #pragma unroll
    for (int r = 0; r < 3; ++r)
        #pragma unroll
        for (int c = 0; c < 3; ++c)
            C[r*3+c] = R[r*3+0]*iK[c] + R[r*3+1]*iK[3+c] + R[r*3+2]*iK[6+c];
    const float tx = trans[bn*3+0], ty = trans[bn*3+1], tz = trans[bn*3+2];
    const float px = post_trans[bn*3+0], py = post_trans[bn*3+1], pz = post_trans[bn*3+2];

    const float xs = (float)w * (703.0f / 43.0f);   // linspace(0, 703, 44)
    const float ys = (float)h * 17.0f;              // linspace(0, 255, 16)

    // p1 = inv(post_rots) @ ([xs,ys,d] - post_trans) = base + col2(iPR)*d
    const float g0 = iPR[0]*(xs-px) + iPR[1]*(ys-py) + iPR[2]*(-pz);
    const float g1 = iPR[3]*(xs-px) + iPR[4]*(ys-py) + iPR[5]*(-pz);
    const float g2 = iPR[6]*(xs-px) + iPR[7]*(ys-py) + iPR[8]*(-pz);

    // ---- per-bin voxel index (lanes = bins; rep0 -> bins 0..31, rep1 -> 32..47)
    int base0, base1;
    #pragma unroll
    for (int rep = 0; rep < 2; ++rep) {
        const int d = lane + rep * 32;
        const float dval = 4.0f + (float)d;         // D_LO + d*D_STEP
        const float p1x = g0 + iPR[2]*dval;
        const float p1y = g1 + iPR[5]*dval;
        const float p1z = g2 + iPR[8]*dval;
        const float qx = p1x * p1z, qy = p1y * p1z, qz = p1z;
        const float X = C[0]*qx + C[1]*qy + C[2]*qz + tx;
        const float Y = C[3]*qx + C[4]*qy + C[5]*qz + ty;
        const float Z = C[6]*qx + C[7]*qy + C[8]*qz + tz;
        // gidx = floor((p - (bx - dx/2)) / dx); bx-dx/2 = (-50,-50,-10); dx = (.5,.5,20)
        const int ix = (int)floorf((X + 50.0f) * 2.0f);
        const int iy = (int)floorf((Y + 50.0f) * 2.0f);
        const int iz = (int)floorf((Z + 10.0f) * 0.05f);
        const bool valid = (ix >= 0) & (ix < NX_) & (iy >= 0) & (iy < NY_) & (iz == 0);
        int base;
        if (WS) {   // acc[cell*64 + c]; invalid bins -> dump cell (cropped by transpose)
            base = (valid ? ((b * NX_ + ix) * NY_ + iy) : CELLS) * 64;
        } else {    // acc[((b*64+c)*NX+ix)*NY+iy]; invalid -> skip
            base = valid ? (b * (CAMC_ * NX_ * NY_) + ix * NY_ + iy) : -1;
        }
        if (rep == 0) base0 = base; else base1 = base;
    }

    // ---- A operands per depth-bin tile (lanes 0..15 carry K=0 column; rest zero)
    const float d16 = __shfl_xor(depth0, 16, 32);   // bins 16..31 into lanes 0..15
    float aK0[3];
    aK0[0] = hi ? 0.0f : depth0;                    // bins  0..15
    aK0[1] = hi ? 0.0f : d16;                       // bins 16..31
    aK0[2] = hi ? 0.0f : depth1;                    // bins 32..47 (41..47 exact 0)

    // ---- lifted = depth (48x1) x feat (1x64) on the matrix core; scatter rows
    #pragma unroll
    for (int t = 0; t < 3; ++t) {
        // bases for the 8 rows this lane's half-wave will scatter: m = t*16 + r + hi*8
        int bArr[8];
        #pragma unroll
        for (int r = 0; r < 8; ++r) {
            const int src = (t < 2) ? (t * 16 + r + (hi ? 8 : 0))   // from base0 lanes 0..31
                                    : (r + (hi ? 8 : 0));           // from base1 lanes 0..15
            bArr[r] = __shfl((t < 2) ? base0 : base1, src, 32);
        }
        v2f a; a[0] = aK0[t]; a[1] = 0.0f;
        #pragma unroll
        for (int u = 0; u < 4; ++u) {
            v2f bb; bb[0] = fb[u]; bb[1] = 0.0f;
            v8f cz = {};
            v8f d8 = __builtin_amdgcn_wmma_f32_16x16x4_f32(
                false, a, false, bb, (short)0, cz, false, false);
            #pragma unroll
            for (int r = 0; r < 8; ++r) {
                if (WS) {
                    const unsigned idx = (unsigned)(bArr[r] + u * 16 + cOff);
                    atomicAdd(acc + idx, d8[r]);            // unconditional, 64B/half-wave
                } else {
                    if (bArr[r] >= 0) {
                        const unsigned idx = (unsigned)(bArr[r] + (u * 16 + cOff) * (NX_ * NY_));
                        atomicAdd(acc + idx, d8[r]);
                    }
                }
            }
        }
    }
}

// ---------------- kernel 2: acc(b,x,y,c) -> out(b,c,x,y) via LDS tile ----------------
__global__ __launch_bounds__(256) void lss_transpose(
    const float* __restrict__ acc, float* __restrict__ out)
{
    __shared__ float tile[NY_ * 65];           // stride 65 -> conflict-free column reads
    const int b = blockIdx.x / NX_;
    const int x = blockIdx.x % NX_;
    const float* src = acc + (long)(b * NX_ + x) * NY_ * 64;
    for (int i = threadIdx.x; i < NY_ * 64; i += 256) {
        const int y = i >> 6, c = i & 63;
        tile[y * 65 + c] = src[i];             // coalesced global read
    }
    __syncthreads();
    float* dst = out + ((long)b * CAMC_ * NX_ + x) * NY_;
    for (int j = threadIdx.x; j < NY_ * 64; j += 256) {
        const int c = j / NY_, y = j % NY_;
        dst[(long)c * NX_ * NY_ + y] = tile[y * 65 + c];  // coalesced global write
    }
}

// ---------------- launcher ----------------
extern "C" void kernel_launch(void* const* d_in, const int* in_sizes, int n_in,
                              void* d_out, int out_size, void* d_ws, size_t ws_size,
                              hipStream_t stream) {
    const float* xf         = (const float*)d_in[0];
    const float* rots       = (const float*)d_in[1];
    const float* trans      = (const float*)d_in[2];
    const float* intrins    = (const float*)d_in[3];
    const float* post_rots  = (const float*)d_in[4];
    const float* post_trans = (const float*)d_in[5];
    float* out = (float*)d_out;

    const size_t wsNeeded = (size_t)(CELLS + 1) * 64 * sizeof(float); // +1 dump cell
    const bool useWs = (ws_size >= wsNeeded);

    if (useWs) {
        float* acc = (float*)d_ws;
        lss_zero<<<2048, 256, 0, stream>>>((float4*)acc, (long)(CELLS + 1) * 64 / 4);
        lss_main<true><<<NPIX / 8, 256, 0, stream>>>(xf, rots, trans, intrins,
                                                     post_rots, post_trans, acc);
        lss_transpose<<<B_ * NX_, 256, 0, stream>>>(acc, out);
    } else {
        lss_zero<<<2048, 256, 0, stream>>>((float4*)out, (long)B_ * CAMC_ * NX_ * NY_ / 4);
        lss_main<false><<<NPIX / 8, 256, 0, stream>>>(xf, rots, trans, intrins,
                                                      post_rots, post_trans, out);
    }
}